// GAT_66640712565429
// MI455X (gfx1250) — compile-verified
//
#include <hip/hip_runtime.h>
#include <hip/hip_bf16.h>

// ---------------------------------------------------------------------------
// GAT (3 layers, 4 heads x 16 ch) for MI455X / gfx1250.
// GEMMs use V_WMMA_F32_16X16X4_F32 (full fp32 matrix pipe, exact vs reference).
// Edge phase: 3 passes with order-preserving-uint atomicMax for segment max,
// f32 atomicAdd for segment sums. All scratch carved from d_ws.
// ---------------------------------------------------------------------------

typedef __attribute__((ext_vector_type(2))) float v2f;
typedef __attribute__((ext_vector_type(8))) float v8f;

#define NEG_SLOPE 0.2f
#define GAT_EPS   1e-16f

// -------------------- fp32 WMMA GEMM: Hout[N,MO] = X[N,K] @ W[K,MO] ---------
// One wave computes a 16-row strip across all MO columns.
// A-matrix 16x4 layout : lane = m%16 + 16*(k>=2), vgpr j -> K = 2*(lane/16)+j
// B-matrix 4x16 layout : lane = n%16 + 16*(k>=2), vgpr j -> K = 2*(lane/16)+j
// C/D 16x16 layout     : vgpr r -> row = r + 8*(lane/16), col = lane%16
template <int KDIM, int MO>
__global__ __launch_bounds__(128) void gemm_wmma_f32(
    const float* __restrict__ X, const float* __restrict__ W,
    float* __restrict__ Hout, int ntiles) {
  const int wave = threadIdx.x >> 5;
  const int tile = blockIdx.x * 4 + wave;
  if (tile >= ntiles) return;                 // wave-uniform: EXEC stays all-1s
  const int lane  = threadIdx.x & 31;
  const int lmod  = lane & 15;
  const int lhalf = lane >> 4;
  const int row0  = tile * 16;

  v8f acc[MO / 16];
#pragma unroll
  for (int t = 0; t < MO / 16; ++t) acc[t] = v8f{0, 0, 0, 0, 0, 0, 0, 0};

  const float* xrow = X + (size_t)(row0 + lmod) * KDIM;
#pragma unroll 4
  for (int k0 = 0; k0 < KDIM; k0 += 4) {
    const int ka = k0 + 2 * lhalf;
    v2f a;
    a.x = xrow[ka];
    a.y = xrow[ka + 1];
#pragma unroll
    for (int t = 0; t < MO / 16; ++t) {
      const float* wr = W + (size_t)ka * MO + t * 16 + lmod;
      v2f b;
      b.x = wr[0];
      b.y = wr[MO];
      acc[t] = __builtin_amdgcn_wmma_f32_16x16x4_f32(
          false, a, false, b, (short)0, acc[t], false, false);
    }
  }
#pragma unroll
  for (int t = 0; t < MO / 16; ++t)
#pragma unroll
    for (int r = 0; r < 8; ++r)
      Hout[(size_t)(row0 + r + 8 * lhalf) * MO + t * 16 + lmod] = acc[t][r];
}

// -------------------- per-node attention logits ----------------------------
// esrc[n,h] = sum_c h[n,h,c]*a_src[h,c]; edst likewise. C fixed = 16.
__global__ void node_attn(const float* __restrict__ Hm,
                          const float* __restrict__ a_src,
                          const float* __restrict__ a_dst,
                          float* __restrict__ esrc, float* __restrict__ edst,
                          int N, int Hh) {
  int idx = blockIdx.x * blockDim.x + threadIdx.x;
  if (idx >= N * Hh) return;
  const int h = idx % Hh;
  const int n = idx / Hh;
  const float* row = Hm + ((size_t)n * Hh + h) * 16;
  float s = 0.f, d = 0.f;
#pragma unroll
  for (int c = 0; c < 16; ++c) {
    const float v = row[c];
    s += v * a_src[h * 16 + c];
    d += v * a_dst[h * 16 + c];
  }
  esrc[idx] = s;
  edst[idx] = d;
}

// -------------------- fills ------------------------------------------------
__global__ void fill_f32(float* __restrict__ p, float v, int n) {
  int i = blockIdx.x * blockDim.x + threadIdx.x;
  if (i < n) p[i] = v;
}
__global__ void fill_u32(unsigned* __restrict__ p, unsigned v, int n) {
  int i = blockIdx.x * blockDim.x + threadIdx.x;
  if (i < n) p[i] = v;
}

// order-preserving float<->uint mapping so atomicMax(u32) == float max
__device__ __forceinline__ unsigned f2o(float f) {
  unsigned u = __float_as_uint(f);
  return (u & 0x80000000u) ? ~u : (u | 0x80000000u);
}
__device__ __forceinline__ float o2f(unsigned u) {
  return __uint_as_float((u & 0x80000000u) ? (u & 0x7FFFFFFFu) : ~u);
}
#define ORDERED_NEG_INF 0x007FFFFFu  // f2o(-inf)

// -------------------- edge pass 1: leaky-relu score + segment max ----------
__global__ void edge_score_max(const int* __restrict__ ei, int E, int N, int Hh,
                               const float* __restrict__ esrc,
                               const float* __restrict__ edst,
                               float* __restrict__ ebuf,
                               unsigned* __restrict__ mmax) {
  long long t = (long long)blockIdx.x * blockDim.x + threadIdx.x;
  const long long Etot = (long long)E + N;
  if (t >= Etot) return;
  int s, d;
  if (t < E) { s = ei[t]; d = ei[(long long)E + t]; }
  else       { s = d = (int)(t - E); }
  for (int h = 0; h < Hh; ++h) {
    float e = esrc[(size_t)s * Hh + h] + edst[(size_t)d * Hh + h];
    e = (e > 0.f) ? e : NEG_SLOPE * e;
    ebuf[t * Hh + h] = e;
    atomicMax(&mmax[(size_t)d * Hh + h], f2o(e));
  }
}

// -------------------- edge pass 2: exp + segment sum -----------------------
__global__ void edge_exp_sum(const int* __restrict__ ei, int E, int N, int Hh,
                             const unsigned* __restrict__ mmax,
                             float* __restrict__ ebuf,
                             float* __restrict__ denom) {
  long long t = (long long)blockIdx.x * blockDim.x + threadIdx.x;
  const long long Etot = (long long)E + N;
  if (t >= Etot) return;
  int d;
  if (t < E) d = ei[(long long)E + t];
  else       d = (int)(t - E);
  for (int h = 0; h < Hh; ++h) {
    const float m  = o2f(mmax[(size_t)d * Hh + h]);
    const float ex = __expf(ebuf[t * Hh + h] - m);
    ebuf[t * Hh + h] = ex;
    atomicAdd(&denom[(size_t)d * Hh + h], ex);
  }
}

// -------------------- edge pass 3: alpha-weighted aggregation --------------
// one thread per (edge, head); 4x float4 gather of h[src], 16 atomic adds.
__global__ void edge_aggregate(const int* __restrict__ ei, int E, int N, int Hh,
                               const float* __restrict__ ebuf,
                               const float* __restrict__ denom,
                               const float* __restrict__ Hm,
                               float* __restrict__ outb) {
  long long idx = (long long)blockIdx.x * blockDim.x + threadIdx.x;
  const long long tot = ((long long)E + N) * Hh;
  if (idx >= tot) return;
  const int h = (int)(idx % Hh);
  const long long t = idx / Hh;
  int s, d;
  if (t < E) { s = ei[t]; d = ei[(long long)E + t]; }
  else       { s = d = (int)(t - E); }
  const float alpha = ebuf[t * Hh + h] / (denom[(size_t)d * Hh + h] + GAT_EPS);
  const float4* hp = (const float4*)(Hm + ((size_t)s * Hh + h) * 16);
  float* op = outb + ((size_t)d * Hh + h) * 16;
  const float4 v0 = hp[0], v1 = hp[1], v2 = hp[2], v3 = hp[3];
  atomicAdd(op + 0,  alpha * v0.x); atomicAdd(op + 1,  alpha * v0.y);
  atomicAdd(op + 2,  alpha * v0.z); atomicAdd(op + 3,  alpha * v0.w);
  atomicAdd(op + 4,  alpha * v1.x); atomicAdd(op + 5,  alpha * v1.y);
  atomicAdd(op + 6,  alpha * v1.z); atomicAdd(op + 7,  alpha * v1.w);
  atomicAdd(op + 8,  alpha * v2.x); atomicAdd(op + 9,  alpha * v2.y);
  atomicAdd(op + 10, alpha * v2.z); atomicAdd(op + 11, alpha * v2.w);
  atomicAdd(op + 12, alpha * v3.x); atomicAdd(op + 13, alpha * v3.y);
  atomicAdd(op + 14, alpha * v3.z); atomicAdd(op + 15, alpha * v3.w);
}

// -------------------- bias (+ optional ReLU) -------------------------------
__global__ void bias_act(const float* __restrict__ in, const float* __restrict__ b,
                         float* __restrict__ out, int N, int M, int relu) {
  int idx = blockIdx.x * blockDim.x + threadIdx.x;
  if (idx >= N * M) return;
  float v = in[idx] + b[idx % M];
  if (relu) v = fmaxf(v, 0.f);
  out[idx] = v;
}

// ---------------------------------------------------------------------------
static inline int cdiv(long long a, int b) { return (int)((a + b - 1) / b); }

struct LayerBufs {
  float* hbuf; float* outb; float* esrc; float* edst;
  unsigned* mmax; float* denom; float* ebuf;
};

static void run_edge_phase(const LayerBufs& L, const int* ei, int E, int N,
                           int Hh, hipStream_t stream) {
  const long long Etot = (long long)E + N;
  const int Mo = Hh * 16;
  fill_u32<<<cdiv(N * Hh, 256), 256, 0, stream>>>(L.mmax, ORDERED_NEG_INF, N * Hh);
  fill_f32<<<cdiv(N * Hh, 256), 256, 0, stream>>>(L.denom, 0.f, N * Hh);
  fill_f32<<<cdiv((long long)N * Mo, 256), 256, 0, stream>>>(L.outb, 0.f, N * Mo);
  edge_score_max<<<cdiv(Etot, 256), 256, 0, stream>>>(ei, E, N, Hh, L.esrc,
                                                      L.edst, L.ebuf, L.mmax);
  edge_exp_sum<<<cdiv(Etot, 256), 256, 0, stream>>>(ei, E, N, Hh, L.mmax,
                                                    L.ebuf, L.denom);
  edge_aggregate<<<cdiv(Etot * Hh, 256), 256, 0, stream>>>(
      ei, E, N, Hh, L.ebuf, L.denom, L.hbuf, L.outb);
}

extern "C" void kernel_launch(void* const* d_in, const int* in_sizes, int n_in,
                              void* d_out, int out_size, void* d_ws, size_t ws_size,
                              hipStream_t stream) {
  const float* x   = (const float*)d_in[0];
  const int*   ei  = (const int*)d_in[1];
  const float* W1  = (const float*)d_in[2];
  const float* a1s = (const float*)d_in[3];
  const float* a1d = (const float*)d_in[4];
  const float* b1  = (const float*)d_in[5];
  const float* W2  = (const float*)d_in[6];
  const float* a2s = (const float*)d_in[7];
  const float* a2d = (const float*)d_in[8];
  const float* b2  = (const float*)d_in[9];
  const float* W3  = (const float*)d_in[10];
  const float* a3s = (const float*)d_in[11];
  const float* a3d = (const float*)d_in[12];
  const float* b3  = (const float*)d_in[13];

  const int B = 4, F_IN = 128, H = 4, HC = 64;
  const int E = in_sizes[1] / 2;
  const int N = in_sizes[0] / (B * F_IN);
  const int NCLS = in_sizes[13];           // 16

  // workspace carve-up (floats)
  float* ws = (float*)d_ws;
  float*    hbuf  = ws;                                  // N*64
  float*    xbuf  = hbuf + (size_t)N * HC;               // N*64
  float*    outb  = xbuf + (size_t)N * HC;               // N*64
  float*    esrc  = outb + (size_t)N * HC;               // N*H
  float*    edst  = esrc + (size_t)N * H;                // N*H
  unsigned* mmax  = (unsigned*)(edst + (size_t)N * H);   // N*H
  float*    denom = (float*)(mmax + (size_t)N * H);      // N*H
  float*    ebuf  = denom + (size_t)N * H;               // (E+N)*H

  const int ntiles = (N + 15) / 16;
  const int gemm_grid = (ntiles + 3) / 4;

  LayerBufs L{hbuf, outb, esrc, edst, mmax, denom, ebuf};

  for (int b = 0; b < B; ++b) {
    // ---- layer 1: 128 -> 4x16, concat, ReLU ----
    gemm_wmma_f32<128, 64><<<gemm_grid, 128, 0, stream>>>(
        x + (size_t)b * N * F_IN, W1, hbuf, ntiles);
    node_attn<<<cdiv((long long)N * H, 256), 256, 0, stream>>>(
        hbuf, a1s, a1d, esrc, edst, N, H);
    run_edge_phase(L, ei, E, N, H, stream);
    bias_act<<<cdiv((long long)N * HC, 256), 256, 0, stream>>>(
        outb, b1, xbuf, N, HC, 1);

    // ---- layer 2: 64 -> 4x16, concat, ReLU ----
    gemm_wmma_f32<64, 64><<<gemm_grid, 128, 0, stream>>>(xbuf, W2, hbuf, ntiles);
    node_attn<<<cdiv((long long)N * H, 256), 256, 0, stream>>>(
        hbuf, a2s, a2d, esrc, edst, N, H);
    run_edge_phase(L, ei, E, N, H, stream);
    bias_act<<<cdiv((long long)N * HC, 256), 256, 0, stream>>>(
        outb, b2, xbuf, N, HC, 1);

    // ---- layer 3: 64 -> 1x16, no ReLU, -> d_out ----
    gemm_wmma_f32<64, 16><<<gemm_grid, 128, 0, stream>>>(xbuf, W3, hbuf, ntiles);
    node_attn<<<cdiv((long long)N * 1, 256), 256, 0, stream>>>(
        hbuf, a3s, a3d, esrc, edst, N, 1);
    run_edge_phase(L, ei, E, N, 1, stream);
    bias_act<<<cdiv((long long)N * NCLS, 256), 256, 0, stream>>>(
        outb, b3, (float*)d_out + (size_t)b * N * NCLS, N, NCLS, 0);
  }
}